// feature_augmenation_74302934221510
// MI455X (gfx1250) — compile-verified
//
#include <hip/hip_runtime.h>
#include <stdint.h>
#include <stddef.h>

#define BB 65536
#define CC 1000
#define DD 512
#define EMA_R 0.1f
#define NBUF 8            // async pipeline depth: 8 x 2KB rows in LDS
#define RTHREADS 128      // 4 waves; each thread owns 4 columns (float4)

// ---------------- CDNA5 async global->LDS copy (gfx1250) --------------------
// Each lane copies 16 bytes from the global address gaddr to the LDS byte address lds_byte. Tracked by ASYNCcnt.
__device__ __forceinline__ void async_load_b128(uint32_t lds_byte, const void* gaddr) {
  asm volatile("global_load_async_to_lds_b128 %0, %1, off"
               :: "v"(lds_byte), "v"(gaddr)
               : "memory");
}

// S_WAIT_ASYNCCNT takes an immediate; select it at runtime via a small switch.
__device__ __forceinline__ void wait_async_le(int n) {
  switch (n) {
    case 0: asm volatile("s_wait_asynccnt 0" ::: "memory"); break;
    case 1: asm volatile("s_wait_asynccnt 1" ::: "memory"); break;
    case 2: asm volatile("s_wait_asynccnt 2" ::: "memory"); break;
    case 3: asm volatile("s_wait_asynccnt 3" ::: "memory"); break;
    case 4: asm volatile("s_wait_asynccnt 4" ::: "memory"); break;
    case 5: asm volatile("s_wait_asynccnt 5" ::: "memory"); break;
    case 6: asm volatile("s_wait_asynccnt 6" ::: "memory"); break;
    default: asm volatile("s_wait_asynccnt 7" ::: "memory"); break;
  }
}

// Make sure our own in-flight LDS reads of a buffer have drained before the
// async engine is allowed to overwrite that buffer slot.
__device__ __forceinline__ void wait_ds_drained() {
  asm volatile("s_wait_dscnt 0" ::: "memory");
}

// ---------------- Phase 0: zero class counters ------------------------------
__global__ void k_zero(int* __restrict__ count) {
  int i = blockIdx.x * blockDim.x + threadIdx.x;
  if (i < CC) count[i] = 0;
}

// ---------------- Phase 1: label histogram ----------------------------------
__global__ void k_hist(const int* __restrict__ labels, int* __restrict__ count) {
  int i = blockIdx.x * blockDim.x + threadIdx.x;
  if (i < BB) atomicAdd(&count[labels[i]], 1);
}

// ---------------- Phase 2: exclusive scan over C=1000 (one block) -----------
__global__ void __launch_bounds__(1024) k_scan(const int* __restrict__ count,
                                               int* __restrict__ start,
                                               int* __restrict__ cursor) {
  __shared__ int tmp[1024];
  const int t = threadIdx.x;
  int v = (t < CC) ? count[t] : 0;
  tmp[t] = v;
  __syncthreads();
  for (int off = 1; off < 1024; off <<= 1) {
    int add = (t >= off) ? tmp[t - off] : 0;
    __syncthreads();
    tmp[t] += add;
    __syncthreads();
  }
  if (t < CC) {
    int excl = tmp[t] - v;
    start[t]  = excl;
    cursor[t] = excl;
  }
}

// ---------------- Phase 3: scatter row ids into class-sorted order ----------
__global__ void k_scatter(const int* __restrict__ labels,
                          int* __restrict__ cursor, int* __restrict__ order) {
  int i = blockIdx.x * blockDim.x + threadIdx.x;
  if (i < BB) {
    int pos = atomicAdd(&cursor[labels[i]], 1);
    order[pos] = i;
  }
}

// ---------------- Phase 4: per-class reduce + fused EMA finalize ------------
// Each lane async-copies 16B of a row into LDS and later reads back exactly
// those 16B, so the pipeline is wave-private: no __syncthreads needed at all.
__global__ void __launch_bounds__(RTHREADS) k_reduce(
    const float* __restrict__ feature,
    const float* __restrict__ fmean,
    const float* __restrict__ fvar,
    const int*   __restrict__ used,
    const int*   __restrict__ count,
    const int*   __restrict__ start,
    const int*   __restrict__ order,
    float*       __restrict__ out) {
  const int c   = blockIdx.x;       // one class per workgroup
  const int tid = threadIdx.x;      // 0..127, owns 4 consecutive columns
  const int n    = count[c];
  const int base = start[c];

  __shared__ alignas(16) float buf[NBUF][DD];   // 16 KB: 8 rows of 2 KB
  const uint32_t lds0   = (uint32_t)(uintptr_t)(void*)&buf[0][0];
  const uint32_t my_off = (uint32_t)tid * 16u;  // 16 B per lane per row

  float s10 = 0.f, s11 = 0.f, s12 = 0.f, s13 = 0.f;   // sum x
  float s20 = 0.f, s21 = 0.f, s22 = 0.f, s23 = 0.f;   // sum x^2

  if (n > 0) {
    // Prefill the async pipeline (up to NBUF rows in flight per wave).
    const int pre = (n < NBUF) ? n : NBUF;
    for (int k = 0; k < pre; ++k) {
      const int r = order[base + k];
      async_load_b128(lds0 + (uint32_t)k * (DD * 4u) + my_off,
                      feature + (size_t)r * DD + (size_t)tid * 4);
    }
    for (int i = 0; i < n; ++i) {
      int inflight = n - i; if (inflight > NBUF) inflight = NBUF;
      wait_async_le(inflight - 1);   // in-order completion => oldest row landed

      const float* row = &buf[i & (NBUF - 1)][tid * 4];
      const float x0 = row[0], x1 = row[1], x2 = row[2], x3 = row[3];
      s10 += x0; s11 += x1; s12 += x2; s13 += x3;
      s20 = fmaf(x0, x0, s20); s21 = fmaf(x1, x1, s21);
      s22 = fmaf(x2, x2, s22); s23 = fmaf(x3, x3, s23);

      const int nxt = i + NBUF;
      if (nxt < n) {
        wait_ds_drained();           // our ds_load of this slot has completed
        const int r = order[base + nxt];
        async_load_b128(lds0 + (uint32_t)(i & (NBUF - 1)) * (DD * 4u) + my_off,
                        feature + (size_t)r * DD + (size_t)tid * 4);
      }
    }
  }

  // ---- fused finalize: EMA mean/var update for this class's 4 columns ----
  const size_t o = (size_t)c * DD + (size_t)tid * 4;
  const float m0 = fmean[o], m1 = fmean[o + 1], m2 = fmean[o + 2], m3 = fmean[o + 3];
  const float v0 = fvar[o],  v1 = fvar[o + 1],  v2 = fvar[o + 2],  v3 = fvar[o + 3];
  const int u = used[c];

  float nm0 = m0, nm1 = m1, nm2 = m2, nm3 = m3;
  float nv0 = v0, nv1 = v1, nv2 = v2, nv3 = v3;

  if (n > 0) {
    const float fn = (float)n;
    const float bm0 = s10 / fn, bm1 = s11 / fn, bm2 = s12 / fn, bm3 = s13 / fn;
    if (u != 0) {
      nm0 = EMA_R * bm0 + (1.0f - EMA_R) * m0;
      nm1 = EMA_R * bm1 + (1.0f - EMA_R) * m1;
      nm2 = EMA_R * bm2 + (1.0f - EMA_R) * m2;
      nm3 = EMA_R * bm3 + (1.0f - EMA_R) * m3;
    } else {
      nm0 = bm0; nm1 = bm1; nm2 = bm2; nm3 = bm3;
    }
    float denom = fn - 1.0f; if (denom < 1.0f) denom = 1.0f;
    const float ss0 = s20 - 2.0f * nm0 * s10 + fn * nm0 * nm0;
    const float ss1 = s21 - 2.0f * nm1 * s11 + fn * nm1 * nm1;
    const float ss2 = s22 - 2.0f * nm2 * s12 + fn * nm2 * nm2;
    const float ss3 = s23 - 2.0f * nm3 * s13 + fn * nm3 * nm3;
    const float bv0 = ss0 / denom, bv1 = ss1 / denom, bv2 = ss2 / denom, bv3 = ss3 / denom;
    if (u != 0) {
      nv0 = EMA_R * bv0 + (1.0f - EMA_R) * v0;
      nv1 = EMA_R * bv1 + (1.0f - EMA_R) * v1;
      nv2 = EMA_R * bv2 + (1.0f - EMA_R) * v2;
      nv3 = EMA_R * bv3 + (1.0f - EMA_R) * v3;
    } else {
      nv0 = bv0; nv1 = bv1; nv2 = bv2; nv3 = bv3;
    }
  }

  float* out_mean = out;
  float* out_var  = out + (size_t)CC * DD;
  float* out_used = out + 2 * (size_t)CC * DD;
  out_mean[o] = nm0; out_mean[o + 1] = nm1; out_mean[o + 2] = nm2; out_mean[o + 3] = nm3;
  out_var[o]  = nv0; out_var[o + 1]  = nv1; out_var[o + 2]  = nv2; out_var[o + 3]  = nv3;
  if (tid == 0) out_used[c] = (float)(u + ((n > 0 && u == 0) ? 1 : 0));
}

// ---------------------------------------------------------------------------
extern "C" void kernel_launch(void* const* d_in, const int* in_sizes, int n_in,
                              void* d_out, int out_size, void* d_ws, size_t ws_size,
                              hipStream_t stream) {
  const float* feature = (const float*)d_in[0];
  const int*   labels  = (const int*)d_in[1];
  const float* fmean   = (const float*)d_in[2];
  const float* fvar    = (const float*)d_in[3];
  const int*   used    = (const int*)d_in[4];
  (void)in_sizes; (void)n_in; (void)out_size; (void)ws_size;

  int* ws     = (int*)d_ws;
  int* count  = ws;          // [CC]  (padded slots of 1024)
  int* start  = ws + 1024;   // [CC]
  int* cursor = ws + 2048;   // [CC]
  int* order  = ws + 3072;   // [BB]

  k_zero   <<<(CC + 255) / 256, 256, 0, stream>>>(count);
  k_hist   <<<BB / 256,         256, 0, stream>>>(labels, count);
  k_scan   <<<1,               1024, 0, stream>>>(count, start, cursor);
  k_scatter<<<BB / 256,         256, 0, stream>>>(labels, cursor, order);
  k_reduce <<<CC,          RTHREADS, 0, stream>>>(feature, fmean, fvar, used,
                                                  count, start, order, (float*)d_out);
}